// NMS_51642686767880
// MI455X (gfx1250) — compile-verified
//
#include <hip/hip_runtime.h>
#include <stdint.h>

// Problem constants (match reference)
#define NC 144      // checks (rows of H)
#define NV 576      // variables (cols of H)
#define NB 1024     // batch
#define NITERS 3
#define RD 64       // row-CSR stride (power of 2 for shift indexing; max row deg ~34)
#define CMAX 8      // column-CSR stride (col weight 4 + slack from the row-fix loop)
#define TPB 256     // 8 wave32s per block

// ---------------------------------------------------------------------------
// Workspace layout (all 16B-aligned):
//   int            rdeg [NC]           @ 0            (576 B)
//   unsigned short rcols[NC*RD]        @ 576          (18432 B)
//   int            cdeg [NV]           @ 19008        (2304 B)
//   unsigned short cent [NV*CMAX]      @ 21312        (9216 B)   cent = c*RD + e
// ---------------------------------------------------------------------------
#define WS_RCOLS 576
#define WS_CDEG  19008
#define WS_CENT  21312

// Build row CSR and column CSR from dense H. Phases are independent (phase 2
// recomputes the in-row edge slot by scanning H), so no intra-kernel sync of
// global data is needed.
__global__ void ldpc_build_csr(const float* __restrict__ H,
                               int* __restrict__ rdeg,
                               unsigned short* __restrict__ rcols,
                               int* __restrict__ cdeg,
                               unsigned short* __restrict__ cent) {
  const int t = threadIdx.x;
  if (t < NC) {                       // row adjacency
    int d = 0;
    for (int v = 0; v < NV; ++v) {
      if (H[t * NV + v] > 0.5f) {
        if (d < RD) rcols[t * RD + d] = (unsigned short)v;
        ++d;
      }
    }
    rdeg[t] = d < RD ? d : RD;
  }
  if (t < NV) {                       // column adjacency -> (row, edge-slot)
    int d = 0;
    for (int c = 0; c < NC; ++c) {
      if (H[c * NV + t] > 0.5f) {
        int e = 0;                    // slot of column t within row c's list
        for (int vv = 0; vv < t; ++vv) e += (H[c * NV + vv] > 0.5f) ? 1 : 0;
        if (e > RD - 1) e = RD - 1;
        if (d < CMAX) cent[t * CMAX + d] = (unsigned short)(c * RD + e);
        ++d;
      }
    }
    cdeg[t] = d < CMAX ? d : CMAX;
  }
}

// Async global->LDS b128 copy of one 16-byte chunk (CDNA5 async-tensor path).
__device__ __forceinline__ void async_b128(const void* gptr, void* ldsptr) {
  unsigned long long g = (unsigned long long)gptr;
  unsigned l = (unsigned)(uintptr_t)ldsptr;   // low 32 bits = LDS offset
  asm volatile("global_load_async_to_lds_b128 %0, %1, off"
               :: "v"(l), "v"(g) : "memory");
}

__global__ __launch_bounds__(TPB) void ldpc_minsum(
    const float* __restrict__ r, const float* __restrict__ alpha,
    const int* __restrict__ grdeg, const unsigned short* __restrict__ grcols,
    const int* __restrict__ gcdeg, const unsigned short* __restrict__ gcent,
    float* __restrict__ out) {
  __shared__ __align__(16) float rbuf[NV];            // this batch row of r
  __shared__ __align__(16) float colv[NV];            // col sums of E
  __shared__ __align__(16) float Medge[NC * RD];      // per-edge M (then E)
  __shared__ __align__(16) unsigned short srcols[NC * RD];
  __shared__ __align__(16) unsigned short scent[NV * CMAX];
  __shared__ int srdeg[NC];
  __shared__ int scdeg[NV];

  const int tid = threadIdx.x;
  const int b = blockIdx.x;

  // ---- async DMA staging into LDS (tracked by ASYNCcnt) ----
  if (tid < NV / 4)                                           // 144 x 16B: r row
    async_b128(r + (size_t)b * NV + tid * 4, &rbuf[tid * 4]);
  for (int t = tid; t < NC * RD / 8; t += TPB)                // row cols (L2 hit)
    async_b128(grcols + t * 8, &srcols[t * 8]);
  for (int t = tid; t < NV * CMAX / 8; t += TPB)              // col entries
    async_b128(gcent + t * 8, &scent[t * 8]);
  for (int t = tid; t < NC; t += TPB) srdeg[t] = grdeg[t];
  for (int t = tid; t < NV; t += TPB) scdeg[t] = gcdeg[t];
  asm volatile("s_wait_asynccnt 0" ::: "memory");
  __syncthreads();

  // ---- init M[c,v] = r[v] on edges ----
  for (int t = tid; t < NC * RD; t += TPB) {
    const int c = t >> 6, e = t & (RD - 1);
    if (e < srdeg[c]) Medge[t] = rbuf[srcols[t]];
  }
  __syncthreads();

#pragma unroll
  for (int it = 0; it < NITERS; ++it) {
    const float a = alpha[it];

    // ---- check-node phase: two-smallest + sign parity, E overwrites M ----
    if (tid < NC) {
      const int base = tid * RD;
      const int d = srdeg[tid];
      float m1 = __builtin_inff(), m2 = __builtin_inff();
      int e1 = 0, negc = 0;
      bool anyz = false;
      for (int e = 0; e < d; ++e) {
        const float m = Medge[base + e];
        const float am = fabsf(m);
        if (am < m1) { m2 = m1; m1 = am; e1 = e; }     // strict '<' keeps
        else if (am < m2) { m2 = am; }                 // first-occurrence argmin
        negc += (m < 0.0f) ? 1 : 0;
        anyz |= (m == 0.0f);
      }
      const float rs = anyz ? 0.0f : ((negc & 1) ? -1.0f : 1.0f);
      for (int e = 0; e < d; ++e) {
        const float m = Medge[base + e];
        const float s = (m > 0.0f) ? 1.0f : ((m < 0.0f) ? -1.0f : 0.0f);
        const float excl = (e == e1) ? m2 : m1;
        Medge[base + e] = a * rs * s * excl;           // E
      }
    }
    __syncthreads();

    // ---- deterministic column sums: col[v] = sum_c E[c,v] ----
    for (int v = tid; v < NV; v += TPB) {
      const int d = scdeg[v];
      float s = 0.0f;
      for (int k = 0; k < d; ++k) s += Medge[scent[v * CMAX + k]];
      colv[v] = s;
    }
    __syncthreads();

    // ---- variable-node update M = col - E + r (not needed after last E) ----
    if (it + 1 < NITERS) {
      for (int t = tid; t < NC * RD; t += TPB) {
        const int c = t >> 6, e = t & (RD - 1);
        if (e < srdeg[c]) {
          const int v = srcols[t];
          Medge[t] = colv[v] - Medge[t] + rbuf[v];
        }
      }
      __syncthreads();
    }
  }

  // ---- output: out = r + col (coalesced) ----
  for (int v = tid; v < NV; v += TPB)
    out[(size_t)b * NV + v] = rbuf[v] + colv[v];
}

extern "C" void kernel_launch(void* const* d_in, const int* in_sizes, int n_in,
                              void* d_out, int out_size, void* d_ws, size_t ws_size,
                              hipStream_t stream) {
  const float* r = (const float*)d_in[0];      // (B, V) f32
  const float* alpha = (const float*)d_in[1];  // (ITERS,) f32
  const float* H = (const float*)d_in[2];      // (C, V) f32
  float* out = (float*)d_out;                  // (B, V) f32

  char* ws = (char*)d_ws;
  int* rdeg = (int*)(ws + 0);
  unsigned short* rcols = (unsigned short*)(ws + WS_RCOLS);
  int* cdeg = (int*)(ws + WS_CDEG);
  unsigned short* cent = (unsigned short*)(ws + WS_CENT);

  ldpc_build_csr<<<1, NV, 0, stream>>>(H, rdeg, rcols, cdeg, cent);
  ldpc_minsum<<<NB, TPB, 0, stream>>>(r, alpha, rdeg, rcols, cdeg, cent, out);
}